// SSI_DDI_Block_27573690040721
// MI455X (gfx1250) — compile-verified
//
#include <hip/hip_runtime.h>
#include <hip/hip_bf16.h>
#include <math.h>

#define Nn 100000
#define Ff 128
#define Hh 2
#define Dd 32
#define HD 64
#define Ee 1600000
#define Gg 2048
#define NEG_SLOPE 0.2f
#define ET (Ee + Nn)   // edges including self-loops

typedef __attribute__((ext_vector_type(2))) float v2f;
typedef __attribute__((ext_vector_type(8))) float v8f;

__device__ __forceinline__ void atomicMaxFloat(float* addr, float val) {
    // float ordering == int ordering for non-negative, reversed-unsigned for negative
    if (val >= 0.0f) atomicMax((int*)addr, __float_as_int(val));
    else             atomicMin((unsigned int*)addr, __float_as_uint(val));
}

__device__ __forceinline__ float leaky(float v) { return v > 0.0f ? v : NEG_SLOPE * v; }

__device__ __forceinline__ void edge_sd(const int* __restrict__ ei, int i, int& s, int& d) {
    if (i < Ee) { s = ei[i]; d = ei[Ee + i]; }
    else        { s = d = i - Ee; }           // appended self-loops
}

// ---------------- K1: xw = node @ W_gat via V_WMMA_F32_16X16X4_F32 ----------------
// One wave computes a 16x64 output tile (4 WMMAs per K-step, K-loop of 128/4).
__global__ void gemm_wmma(const float* __restrict__ A,   // [N,128]
                          const float* __restrict__ B,   // [128,64]
                          float* __restrict__ C) {       // [N,64]
    const int lane = threadIdx.x & 31;
    const int wid  = threadIdx.x >> 5;
    const int tile = blockIdx.x * (blockDim.x >> 5) + wid;   // wave-uniform
    if (tile * 16 >= Nn) return;                              // uniform exit, EXEC stays full
    const int m  = tile * 16 + (lane & 15);
    const int kg = (lane >> 4) * 2;     // K sub-offset: 0 (lanes 0-15) / 2 (lanes 16-31)
    const int nc = lane & 15;           // column within 16-wide tile
    v8f acc0 = {}, acc1 = {}, acc2 = {}, acc3 = {};
    for (int k = 0; k < Ff; k += 4) {
        v2f a, b0, b1, b2, b3;
        const float* Ap = A + (size_t)m * Ff + k + kg;
        a.x = Ap[0]; a.y = Ap[1];
        const float* B0 = B + (size_t)(k + kg) * HD + nc;   // row k+kg
        const float* B1 = B0 + HD;                           // row k+kg+1
        b0.x = B0[0];  b0.y = B1[0];
        b1.x = B0[16]; b1.y = B1[16];
        b2.x = B0[32]; b2.y = B1[32];
        b3.x = B0[48]; b3.y = B1[48];
        acc0 = __builtin_amdgcn_wmma_f32_16x16x4_f32(false, a, false, b0, (short)0, acc0, false, false);
        acc1 = __builtin_amdgcn_wmma_f32_16x16x4_f32(false, a, false, b1, (short)0, acc1, false, false);
        acc2 = __builtin_amdgcn_wmma_f32_16x16x4_f32(false, a, false, b2, (short)0, acc2, false, false);
        acc3 = __builtin_amdgcn_wmma_f32_16x16x4_f32(false, a, false, b3, (short)0, acc3, false, false);
    }
    const int rb = tile * 16 + (lane >> 4) * 8;   // M = rb + r
    #pragma unroll
    for (int r = 0; r < 8; ++r) {
        float* Cp = C + (size_t)(rb + r) * HD + nc;
        Cp[0] = acc0[r]; Cp[16] = acc1[r]; Cp[32] = acc2[r]; Cp[48] = acc3[r];
    }
}

// ---------------- K2: a_src/a_dst = einsum('nhd,hd->nh') ----------------
__global__ void attn_proj(const float* __restrict__ xw,
                          const float* __restrict__ att_src,
                          const float* __restrict__ att_dst,
                          float* __restrict__ as_, float* __restrict__ ad_) {
    int i = blockIdx.x * blockDim.x + threadIdx.x;   // (node, head)
    if (i >= Nn * Hh) return;
    int h = i & 1;
    const float* row = xw + (size_t)(i >> 1) * HD + h * Dd;
    float s = 0.f, d = 0.f;
    #pragma unroll
    for (int k = 0; k < Dd; ++k) {
        float v = row[k];
        s += v * att_src[h * Dd + k];
        d += v * att_dst[h * Dd + k];
    }
    as_[i] = s; ad_[i] = d;
}

// ---------------- K3: init accumulators (d_out is poisoned each run) ----------------
__global__ void init_bufs(float* __restrict__ mx, float* __restrict__ sx,
                          float* __restrict__ sc, float* __restrict__ m2,
                          float* __restrict__ s2, const float* __restrict__ bias,
                          float* __restrict__ x, float* __restrict__ emb) {
    int i = blockIdx.x * blockDim.x + threadIdx.x;
    if (i < Nn * Hh) { mx[i] = -__builtin_inff(); sx[i] = 0.f; }
    if (i < Nn)        sc[i] = 0.f;
    if (i < Gg)      { m2[i] = -__builtin_inff(); s2[i] = 0.f; }
    if (i < Nn * HD)   x[i]  = bias[i & (HD - 1)];
    if (i < Gg * HD)   emb[i] = 0.f;
}

// ---------------- K4: segment max of leaky logits over dst ----------------
__global__ void edge_max(const int* __restrict__ ei, const float* __restrict__ as_,
                         const float* __restrict__ ad_, float* __restrict__ mx) {
    int i = blockIdx.x * blockDim.x + threadIdx.x;
    if (i >= ET) return;
    int s, d; edge_sd(ei, i, s, d);
    #pragma unroll
    for (int h = 0; h < Hh; ++h) {
        float lg = leaky(as_[s * Hh + h] + ad_[d * Hh + h]);
        atomicMaxFloat(&mx[d * Hh + h], lg);
    }
}

// ---------------- K5: segment exp-sum ----------------
__global__ void edge_expsum(const int* __restrict__ ei, const float* __restrict__ as_,
                            const float* __restrict__ ad_, const float* __restrict__ mx,
                            float* __restrict__ sx) {
    int i = blockIdx.x * blockDim.x + threadIdx.x;
    if (i >= ET) return;
    int s, d; edge_sd(ei, i, s, d);
    #pragma unroll
    for (int h = 0; h < Hh; ++h) {
        float lg = leaky(as_[s * Hh + h] + ad_[d * Hh + h]);
        atomicAdd(&sx[d * Hh + h], __expf(lg - mx[d * Hh + h]));
    }
}

// ---------------- K6: weighted message scatter, one wave32 per edge ----------------
__global__ void edge_msg(const int* __restrict__ ei, const float* __restrict__ as_,
                         const float* __restrict__ ad_, const float* __restrict__ mx,
                         const float* __restrict__ sx, const float* __restrict__ xw,
                         float* __restrict__ x) {
    int e = blockIdx.x * (blockDim.x >> 5) + (threadIdx.x >> 5);
    if (e >= ET) return;
    int lane = threadIdx.x & 31;
    int s, d; edge_sd(ei, e, s, d);
    float al0 = __expf(leaky(as_[s * 2 + 0] + ad_[d * 2 + 0]) - mx[d * 2 + 0]) / sx[d * 2 + 0];
    float al1 = __expf(leaky(as_[s * 2 + 1] + ad_[d * 2 + 1]) - mx[d * 2 + 1]) / sx[d * 2 + 1];
    float v0 = xw[(size_t)s * HD + lane]      * al0;   // head 0 cols 0..31
    float v1 = xw[(size_t)s * HD + 32 + lane] * al1;   // head 1 cols 32..63
    atomicAdd(&x[(size_t)d * HD + lane],      v0);
    atomicAdd(&x[(size_t)d * HD + 32 + lane], v1);
}

// ---------------- K7a: per-node 64-wide dot (wave32 per node) ----------------
__global__ void node_dot(const float* __restrict__ x, const float* __restrict__ w,
                         float* __restrict__ out) {
    int n = blockIdx.x * (blockDim.x >> 5) + (threadIdx.x >> 5);
    if (n >= Nn) return;
    int lane = threadIdx.x & 31;
    float p = x[(size_t)n * HD + lane] * w[lane] + x[(size_t)n * HD + 32 + lane] * w[32 + lane];
    #pragma unroll
    for (int off = 16; off > 0; off >>= 1) p += __shfl_down(p, off);
    if (lane == 0) out[n] = p;
}

// ---------------- K7b: scalar relation scatter (GraphConv agg @ W_rel, linearized) ----------------
__global__ void rel_scatter(const int* __restrict__ ei, const float* __restrict__ y,
                            float* __restrict__ sc) {
    int i = blockIdx.x * blockDim.x + threadIdx.x;
    if (i >= Ee) return;                  // original edges only (no self-loops here)
    atomicAdd(&sc[ei[Ee + i]], y[ei[i]]);
}

// ---------------- K8: raw score + per-graph max ----------------
__global__ void score_raw_k(const float* __restrict__ sc, const float* __restrict__ rr,
                            const float* __restrict__ b_rel, const int* __restrict__ batch,
                            float* __restrict__ sraw, float* __restrict__ m2) {
    int i = blockIdx.x * blockDim.x + threadIdx.x;
    if (i >= Nn) return;
    float v = sc[i] + rr[i] + b_rel[0];
    sraw[i] = v;
    atomicMaxFloat(&m2[batch[i]], v);
}

// ---------------- K9: per-graph exp-sum ----------------
__global__ void score_exp(const int* __restrict__ batch, const float* __restrict__ m2,
                          float* __restrict__ sraw, float* __restrict__ s2) {
    int i = blockIdx.x * blockDim.x + threadIdx.x;
    if (i >= Nn) return;
    float e = __expf(sraw[i] - m2[batch[i]]);
    sraw[i] = e;
    atomicAdd(&s2[batch[i]], e);
}

// ---------------- K10: att_x = x*score; global_add_pool (wave32 per node) ----------------
__global__ void pool_k(const float* __restrict__ x, const float* __restrict__ sraw,
                       const float* __restrict__ s2, const int* __restrict__ batch,
                       float* __restrict__ emb) {
    int n = blockIdx.x * (blockDim.x >> 5) + (threadIdx.x >> 5);
    if (n >= Nn) return;
    int lane = threadIdx.x & 31;
    int b = batch[n];
    float sf = sraw[n] / s2[b];
    atomicAdd(&emb[(size_t)b * HD + lane],      x[(size_t)n * HD + lane]      * sf);
    atomicAdd(&emb[(size_t)b * HD + 32 + lane], x[(size_t)n * HD + 32 + lane] * sf);
}

extern "C" void kernel_launch(void* const* d_in, const int* in_sizes, int n_in,
                              void* d_out, int out_size, void* d_ws, size_t ws_size,
                              hipStream_t stream) {
    (void)in_sizes; (void)n_in; (void)out_size; (void)ws_size;
    const float* node    = (const float*)d_in[0];
    const int*   ei      = (const int*)  d_in[1];   // [2,E]: row0 src, row1 dst
    const int*   batch   = (const int*)  d_in[2];
    const float* W_gat   = (const float*)d_in[3];
    const float* att_src = (const float*)d_in[4];
    const float* att_dst = (const float*)d_in[5];
    const float* bias    = (const float*)d_in[6];
    const float* W_rel   = (const float*)d_in[7];
    const float* b_rel   = (const float*)d_in[8];
    const float* W_root  = (const float*)d_in[9];

    float* x   = (float*)d_out;               // [N,64]
    float* emb = x + (size_t)Nn * HD;         // [G,64]

    float* ws   = (float*)d_ws;
    float* xw   = ws;  ws += (size_t)Nn * HD;  // 25.6 MB
    float* as_  = ws;  ws += Nn * Hh;
    float* ad_  = ws;  ws += Nn * Hh;
    float* mx   = ws;  ws += Nn * Hh;
    float* sx   = ws;  ws += Nn * Hh;
    float* y    = ws;  ws += Nn;
    float* rr   = ws;  ws += Nn;
    float* sc   = ws;  ws += Nn;
    float* sraw = ws;  ws += Nn;
    float* m2   = ws;  ws += Gg;
    float* s2   = ws;  ws += Gg;

    // K1: GEMM (6250 16-row tiles, 8 waves/block)
    gemm_wmma<<<(Nn / 16 + 7) / 8, 256, 0, stream>>>(node, W_gat, xw);
    // K2: attention projections
    attn_proj<<<(Nn * Hh + 255) / 256, 256, 0, stream>>>(xw, att_src, att_dst, as_, ad_);
    // K3: init all accumulators (covers every buffer; largest is N*HD)
    init_bufs<<<((size_t)Nn * HD + 255) / 256, 256, 0, stream>>>(mx, sx, sc, m2, s2, bias, x, emb);
    // K4/K5: edge softmax (max, then exp-sum)
    edge_max<<<(ET + 255) / 256, 256, 0, stream>>>(ei, as_, ad_, mx);
    edge_expsum<<<(ET + 255) / 256, 256, 0, stream>>>(ei, as_, ad_, mx, sx);
    // K6: weighted message scatter (wave per edge)
    edge_msg<<<(ET + 7) / 8, 256, 0, stream>>>(ei, as_, ad_, mx, sx, xw, x);
    // K7: y = x@W_rel, rr = x@W_root ; scalar scatter of y over edges
    node_dot<<<(Nn + 7) / 8, 256, 0, stream>>>(x, W_rel, y);
    node_dot<<<(Nn + 7) / 8, 256, 0, stream>>>(x, W_root, rr);
    rel_scatter<<<(Ee + 255) / 256, 256, 0, stream>>>(ei, y, sc);
    // K8-K10: per-graph softmax + weighted pool
    score_raw_k<<<(Nn + 255) / 256, 256, 0, stream>>>(sc, rr, b_rel, batch, sraw, m2);
    score_exp<<<(Nn + 255) / 256, 256, 0, stream>>>(batch, m2, sraw, s2);
    pool_k<<<(Nn + 7) / 8, 256, 0, stream>>>(x, sraw, s2, batch, emb);
}